// TransformerEncoder_70695161692418
// MI455X (gfx1250) — compile-verified
//
#include <hip/hip_runtime.h>
#include <hip/hip_bf16.h>
#include <math.h>
#include <stdint.h>

// ---------------------------------------------------------------------------
// CDNA5 (gfx1250) transformer encoder.
//  - All GEMMs on V_WMMA_F32_16X16X4_F32 (full-f32 path, matches reference).
//  - Weight/activation tiles staged to LDS with GLOBAL_LOAD_ASYNC_TO_LDS_B128
//    (ASYNCcnt-tracked DMA, double buffered, s_wait_asynccnt pipelining).
// ---------------------------------------------------------------------------

typedef __attribute__((ext_vector_type(2))) float v2f;
typedef __attribute__((ext_vector_type(8))) float v8f;

#define B_   256
#define L_   100
#define D_   512
#define F_   2048
#define NL_  4
#define NH_  8
#define HD_  64
#define M_   (B_ * L_)     // 25600 activation rows
#define LP_  112           // L padded to a multiple of 16

#define SA_STRIDE 36       // 64x32 A tile, padded row (16B-aligned, bank-clean)
#define SW_STRIDE 72       // 32x64 W tile, padded row (lane-halves hit disjoint banks)

__device__ __forceinline__ v8f wmma_f32(v2f a, v2f b, v8f c) {
  return __builtin_amdgcn_wmma_f32_16x16x4_f32(
      /*neg_a=*/false, a, /*neg_b=*/false, b,
      /*c_mod=*/(short)0, c, /*reuse_a=*/false, /*reuse_b=*/false);
}

// Async DMA: 16 bytes global -> LDS, tracked by ASYNCcnt (no VGPR staging).
__device__ __forceinline__ void async_copy16(uint32_t lds_off, const float* gptr) {
  asm volatile("global_load_async_to_lds_b128 %0, %1, off"
               :: "v"(lds_off), "v"((uint64_t)(uintptr_t)gptr)
               : "memory");
}

__device__ __forceinline__ uint32_t lds_addr(const void* p) {
  return (uint32_t)(uintptr_t)p;   // low 32 bits of the shared aperture = LDS offset
}

// ---------------------------------------------------------------------------
// x <- 2*x + PE(l, d)   (reference: x + (x + pe))
// ---------------------------------------------------------------------------
__global__ void pos_embed_kernel(const float* __restrict__ x,
                                 float* __restrict__ out) {
  size_t idx = (size_t)blockIdx.x * blockDim.x + threadIdx.x;
  int d = (int)(idx & (D_ - 1));
  int l = (int)((idx / D_) % L_);
  float div = __expf((float)(d & ~1) * (-9.210340372f / (float)D_));
  float ang = (float)l * div;
  float pe  = (d & 1) ? __cosf(ang) : __sinf(ang);
  out[idx] = 2.0f * x[idx] + pe;
}

// ---------------------------------------------------------------------------
// C[M,N] = A[M,K] @ W[K,N] (+bias) (+relu).  Block tile 64(M) x 64(N),
// 8 waves, each wave 16x32 via two accumulators sharing the A fragment.
// K consumed in 32-wide stages: A(64x32) and W(32x64) tiles are DMA'd to LDS
// with async b128 copies, double-buffered; fragments then come from LDS.
// ---------------------------------------------------------------------------
template <bool HAS_BIAS, bool DO_RELU>
__global__ __launch_bounds__(256) void gemm_kernel(
    const float* __restrict__ A, const float* __restrict__ W,
    const float* __restrict__ bias, float* __restrict__ C,
    int M, int N, int K) {
  __shared__ float sA[2][64 * SA_STRIDE];   // 18.4 KB
  __shared__ float sW[2][32 * SW_STRIDE];   // 18.4 KB

  const int tid  = threadIdx.x;
  const int lane = tid & 31;
  const int wave = tid >> 5;
  const int r    = lane & 15;
  const int hh   = lane >> 4;           // K-half of the fragment
  const int wm   = wave & 3;            // 4 waves along M
  const int wn   = wave >> 2;           // 2 waves along N
  const int m0b  = blockIdx.y * 64;
  const int n0b  = blockIdx.x * 64;

  // staging assignment: 4 x b128 per thread per stage (2 for A, 2 for W)
  const int ar0 = tid >> 3,          akg0 = (tid & 7) * 4;          // chunk tid
  const int ar1 = (tid + 256) >> 3,  akg1 = (tid & 7) * 4;          // chunk tid+256
  const int wr0 = tid >> 4,          wcg0 = (tid & 15) * 4;
  const int wr1 = (tid + 256) >> 4,  wcg1 = (tid & 15) * 4;

  v8f acc0 = {};
  v8f acc1 = {};
  const int c0l = wn * 32 + r;          // local cols inside the 64-wide tile
  const int c1l = c0l + 16;

  const int nkb = K >> 5;               // number of 32-wide K stages

  // ---- prologue: stage K-block 0 into buffer 0 ----
  {
    async_copy16(lds_addr(&sA[0][ar0 * SA_STRIDE + akg0]),
                 A + (size_t)(m0b + ar0) * K + akg0);
    async_copy16(lds_addr(&sA[0][ar1 * SA_STRIDE + akg1]),
                 A + (size_t)(m0b + ar1) * K + akg1);
    async_copy16(lds_addr(&sW[0][wr0 * SW_STRIDE + wcg0]),
                 W + (size_t)wr0 * N + n0b + wcg0);
    async_copy16(lds_addr(&sW[0][wr1 * SW_STRIDE + wcg1]),
                 W + (size_t)wr1 * N + n0b + wcg1);
  }

  for (int kb = 0; kb < nkb; ++kb) {
    const int buf = kb & 1;
    if (kb + 1 < nkb) {
      const int nb = buf ^ 1;
      const int kn = (kb + 1) << 5;
      async_copy16(lds_addr(&sA[nb][ar0 * SA_STRIDE + akg0]),
                   A + (size_t)(m0b + ar0) * K + kn + akg0);
      async_copy16(lds_addr(&sA[nb][ar1 * SA_STRIDE + akg1]),
                   A + (size_t)(m0b + ar1) * K + kn + akg1);
      async_copy16(lds_addr(&sW[nb][wr0 * SW_STRIDE + wcg0]),
                   W + (size_t)(kn + wr0) * N + n0b + wcg0);
      async_copy16(lds_addr(&sW[nb][wr1 * SW_STRIDE + wcg1]),
                   W + (size_t)(kn + wr1) * N + n0b + wcg1);
      // 4 copies of the NEXT stage may stay in flight; current stage is done.
      asm volatile("s_wait_asynccnt 4" ::: "memory");
    } else {
      asm volatile("s_wait_asynccnt 0" ::: "memory");
    }
    __syncthreads();   // publish this wave's landed tiles to all 8 waves

    const float* aBase = &sA[buf][(wm * 16 + r) * SA_STRIDE + 2 * hh];
    const float* wBase = &sW[buf][(2 * hh) * SW_STRIDE];
#pragma unroll
    for (int kk = 0; kk < 32; kk += 4) {
      v2f a = *(const v2f*)(aBase + kk);                 // ds_load_b64
      const float* wrow = wBase + kk * SW_STRIDE;
      v2f b0; b0.x = wrow[c0l]; b0.y = wrow[SW_STRIDE + c0l];
      v2f b1; b1.x = wrow[c1l]; b1.y = wrow[SW_STRIDE + c1l];
      acc0 = wmma_f32(a, b0, acc0);
      acc1 = wmma_f32(a, b1, acc1);
    }
    __syncthreads();   // done reading buf before it is overwritten
  }

  const int c0 = n0b + c0l;
  const int c1 = n0b + c1l;
  float bias0 = HAS_BIAS ? bias[c0] : 0.0f;
  float bias1 = HAS_BIAS ? bias[c1] : 0.0f;
  const int mr = m0b + wm * 16 + 8 * hh;
#pragma unroll
  for (int v = 0; v < 8; ++v) {
    float x0 = acc0[v] + bias0;
    float x1 = acc1[v] + bias1;
    if (DO_RELU) { x0 = fmaxf(x0, 0.0f); x1 = fmaxf(x1, 0.0f); }
    C[(size_t)(mr + v) * N + c0] = x0;
    C[(size_t)(mr + v) * N + c1] = x1;
  }
}

// ---------------------------------------------------------------------------
// Attention: one block per (batch, head).  qkv rows are [B*L, 3*D];
// q at col head*64, k at 512+head*64, v at 1024+head*64.
// S = Q K^T (WMMA -> 112x112 LDS) -> masked softmax -> O = P V (WMMA).
// ---------------------------------------------------------------------------
__global__ __launch_bounds__(256) void attn_kernel(
    const float* __restrict__ qkv, const int* __restrict__ mask,
    float* __restrict__ out) {
  __shared__ float sS[LP_ * LP_];      // 50176 B
  __shared__ int   sMask[LP_];

  const int tid  = threadIdx.x;
  const int lane = tid & 31;
  const int wave = tid >> 5;
  const int r    = lane & 15;
  const int hh   = lane >> 4;
  const int b    = blockIdx.x >> 3;
  const int head = blockIdx.x & 7;
  const size_t base = (size_t)b * L_ * (3 * D_);

  if (tid < L_)       sMask[tid] = mask[b * L_ + tid];
  else if (tid < LP_) sMask[tid] = 0;

  // ---- S = Q K^T : 7x7 tiles of 16x16, K-dim = 64 ----
  const size_t qoff = base + (size_t)head * HD_ + 2 * hh;
  for (int t = wave; t < 49; t += 8) {
    const int m0 = (t / 7) * 16;
    const int n0 = (t % 7) * 16;
    v8f acc = {};
    const float* qp = qkv + qoff + (size_t)(m0 + r) * (3 * D_);
    const float* kp = qkv + qoff + D_ + (size_t)(n0 + r) * (3 * D_);
#pragma unroll 4
    for (int kk = 0; kk < HD_; kk += 4) {
      v2f a  = *(const v2f*)(qp + kk);
      v2f bb = *(const v2f*)(kp + kk);
      acc = wmma_f32(a, bb, acc);
    }
    const int mr = m0 + 8 * hh;
#pragma unroll
    for (int v = 0; v < 8; ++v) sS[(mr + v) * LP_ + n0 + r] = acc[v];
  }
  __syncthreads();

  // ---- masked softmax over rows (scale = 1, as in reference) ----
  if (tid < L_) {
    float* row = sS + tid * LP_;
    float mx = -INFINITY;
    for (int c = 0; c < L_; ++c)
      if (sMask[c]) mx = fmaxf(mx, row[c]);
    float sum = 0.0f;
    for (int c = 0; c < L_; ++c) {
      float e = sMask[c] ? __expf(row[c] - mx) : 0.0f;
      row[c] = e;
      sum += e;
    }
    float inv = 1.0f / sum;
    for (int c = 0; c < L_; ++c)   row[c] *= inv;
    for (int c = L_; c < LP_; ++c) row[c] = 0.0f;   // zero padded K-dim
  }
  __syncthreads();

  // ---- O = P V : 7(M) x 4(N) tiles, K-dim = 112 (padded cols are 0) ----
  const size_t voff = base + 2 * D_ + (size_t)head * HD_;
  for (int t = wave; t < 28; t += 8) {
    const int m0 = (t >> 2) * 16;
    const int n0 = (t & 3) * 16;
    v8f acc = {};
    for (int kk = 0; kk < LP_; kk += 4) {
      v2f a = *(const v2f*)(sS + (m0 + r) * LP_ + kk + 2 * hh);
      const int row0 = kk + 2 * hh;
      v2f bb;
      bb.x = (row0     < L_) ? qkv[voff + (size_t)row0       * (3 * D_) + n0 + r] : 0.0f;
      bb.y = (row0 + 1 < L_) ? qkv[voff + (size_t)(row0 + 1) * (3 * D_) + n0 + r] : 0.0f;
      acc = wmma_f32(a, bb, acc);
    }
    const int mr = m0 + 8 * hh;
#pragma unroll
    for (int v = 0; v < 8; ++v) {
      const int l = mr + v;
      if (l < L_)
        out[((size_t)b * L_ + l) * D_ + head * HD_ + n0 + r] = acc[v];
    }
  }
}

// ---------------------------------------------------------------------------
// out = LayerNorm(res + y) * g + b   (one block per row of D=512)
// ---------------------------------------------------------------------------
__global__ __launch_bounds__(256) void add_ln_kernel(
    const float* __restrict__ res, const float* __restrict__ y,
    const float* __restrict__ g, const float* __restrict__ be,
    float* __restrict__ out) {
  __shared__ float sSum[256];
  __shared__ float sSq[256];
  const int row = blockIdx.x;
  const int tid = threadIdx.x;
  const size_t off = (size_t)row * D_;

  float v0 = res[off + tid]       + y[off + tid];
  float v1 = res[off + 256 + tid] + y[off + 256 + tid];
  sSum[tid] = v0 + v1;
  sSq[tid]  = v0 * v0 + v1 * v1;
  __syncthreads();
#pragma unroll
  for (int o = 128; o > 0; o >>= 1) {
    if (tid < o) { sSum[tid] += sSum[tid + o]; sSq[tid] += sSq[tid + o]; }
    __syncthreads();
  }
  const float mean = sSum[0] * (1.0f / D_);
  const float var  = sSq[0] * (1.0f / D_) - mean * mean;
  const float inv  = rsqrtf(var + 1e-5f);
  out[off + tid]       = (v0 - mean) * inv * g[tid]       + be[tid];
  out[off + 256 + tid] = (v1 - mean) * inv * g[256 + tid] + be[256 + tid];
}

// ---------------------------------------------------------------------------
// Driver.  Workspace: wsA = [M, F] (qkv / ffn hidden), wsB = [M, D].
// Activations live in d_out for the whole network.
// ---------------------------------------------------------------------------
extern "C" void kernel_launch(void* const* d_in, const int* in_sizes, int n_in,
                              void* d_out, int out_size, void* d_ws,
                              size_t ws_size, hipStream_t stream) {
  (void)in_sizes; (void)n_in; (void)out_size; (void)ws_size;
  const float* x     = (const float*)d_in[0];
  const int*   mask  = (const int*)  d_in[1];
  const float* qkv_w = (const float*)d_in[2];
  const float* fc_w  = (const float*)d_in[3];
  const float* fc_b  = (const float*)d_in[4];
  const float* ln1_g = (const float*)d_in[5];
  const float* ln1_b = (const float*)d_in[6];
  const float* ln2_g = (const float*)d_in[7];
  const float* ln2_b = (const float*)d_in[8];
  const float* w1    = (const float*)d_in[9];
  const float* b1    = (const float*)d_in[10];
  const float* w2    = (const float*)d_in[11];
  const float* b2    = (const float*)d_in[12];

  float* X   = (float*)d_out;                  // [M, D] activations
  float* wsA = (float*)d_ws;                   // [M, F] qkv / ffn hidden
  float* wsB = wsA + (size_t)M_ * F_;          // [M, D] attn out / ffn2 out

  pos_embed_kernel<<<dim3((M_ * D_) / 256), 256, 0, stream>>>(x, X);

  for (int i = 0; i < NL_; ++i) {
    // qkv = X @ qkv_w[i]                       [M, 1536]
    gemm_kernel<false, false><<<dim3((3 * D_) / 64, M_ / 64), 256, 0, stream>>>(
        X, qkv_w + (size_t)i * D_ * 3 * D_, nullptr, wsA, M_, 3 * D_, D_);
    // attention                                [M, 512]
    attn_kernel<<<dim3(B_ * NH_), 256, 0, stream>>>(wsA, mask, wsB);
    // proj = attn @ fc_w[i] + fc_b[i]          [M, 512]
    gemm_kernel<true, false><<<dim3(D_ / 64, M_ / 64), 256, 0, stream>>>(
        wsB, fc_w + (size_t)i * D_ * D_, fc_b + (size_t)i * D_, wsA, M_, D_, D_);
    // X = LN(proj + X)
    add_ln_kernel<<<dim3(M_), 256, 0, stream>>>(
        X, wsA, ln1_g + (size_t)i * D_, ln1_b + (size_t)i * D_, X);
    // h = relu(X @ w1[i] + b1[i])              [M, 2048]
    gemm_kernel<true, true><<<dim3(F_ / 64, M_ / 64), 256, 0, stream>>>(
        X, w1 + (size_t)i * D_ * F_, b1 + (size_t)i * F_, wsA, M_, F_, D_);
    // y2 = h @ w2[i] + b2[i]                   [M, 512]
    gemm_kernel<true, false><<<dim3(D_ / 64, M_ / 64), 256, 0, stream>>>(
        wsA, w2 + (size_t)i * F_ * D_, b2 + (size_t)i * D_, wsB, M_, D_, F_);
    // X = LN(X + y2)
    add_ln_kernel<<<dim3(M_), 256, 0, stream>>>(
        X, wsB, ln2_g + (size_t)i * D_, ln2_b + (size_t)i * D_, X);
  }
}